// ClusterLoss_78907139162644
// MI455X (gfx1250) — compile-verified
//
#include <hip/hip_runtime.h>

typedef __attribute__((ext_vector_type(2))) float v2f;
typedef __attribute__((ext_vector_type(8))) float v8f;

#define HH 1024
#define WW 1024
#define NPLANE (HH * WW)
#define NB 4
#define NC 6
#define EPSF 1e-9f
#define CHUNKS 256            // reduction chunks per batch
#define PIX (NPLANE / CHUNKS) // 4096 pixels per chunk

static __device__ __forceinline__ v8f wmma4(v2f a, v2f b, v8f c) {
  // V_WMMA_F32_16X16X4_F32 : D = A(16x4) x B(4x16) + C, exact f32
  return __builtin_amdgcn_wmma_f32_16x16x4_f32(false, a, false, b, (short)0, c,
                                               false, false);
}

// ---------------------------------------------------------------------------
// Pass 1: horizontal 9-tap box filter of {mask, b*mask, b^2*mask}.
// One wave computes one 16x16 tile: Out(16x16) = A(16x24) x Band(24x16),
// split into six 16x16x4 WMMAs. Band[t][j] = 1 iff j <= t <= j+8.
// ---------------------------------------------------------------------------
__global__ __launch_bounds__(256) void hbox_kernel(
    const float* __restrict__ I, const float* __restrict__ B,
    float* __restrict__ hm, float* __restrict__ hbm, float* __restrict__ hb2m) {
  const int wave = threadIdx.x >> 5;
  const int lane = threadIdx.x & 31;
  const long tile = (long)blockIdx.x * 8 + wave; // 4*64*64 = 16384 tiles
  const int plane = (int)(tile >> 12);
  const int trem = (int)(tile & 4095);
  const int row0 = (trem >> 6) * 16;
  const int x0 = (trem & 63) * 16;
  const long pbase = (long)plane * NPLANE;

  const int mrow = lane & 15;          // A: M index
  const int khalf = (lane >> 4) * 2;   // A/B: K base within chunk
  const int coln = lane & 15;          // B: N index

  v8f am = {}, abm = {}, ab2m = {};
#pragma unroll
  for (int c = 0; c < 6; ++c) {
    v2f a_m, a_bm, a_b2m, bb;
#pragma unroll
    for (int s = 0; s < 2; ++s) {
      const int x = x0 - 4 + 4 * c + khalf + s; // padded window position
      float iv = 0.f, bv = 0.f;
      if (x >= 0 && x < WW) {
        const long off = pbase + (long)(row0 + mrow) * WW + x;
        iv = I[off];
        bv = B[off];
      }
      const float m = iv > 0.f ? 1.f : 0.f;
      a_m[s] = m;
      a_bm[s] = bv * m;
      a_b2m[s] = bv * bv * m;
      const int t = 4 * c + khalf + s;
      bb[s] = (t >= coln && t <= coln + 8) ? 1.f : 0.f;
    }
    am = wmma4(a_m, bb, am);
    abm = wmma4(a_bm, bb, abm);
    ab2m = wmma4(a_b2m, bb, ab2m);
  }

  const int rofs = (lane >> 4) * 8; // C/D layout: lanes 16-31 hold M+8
#pragma unroll
  for (int r = 0; r < 8; ++r) {
    const long off = pbase + (long)(row0 + r + rofs) * WW + (x0 + coln);
    hm[off] = am[r];
    hbm[off] = abm[r];
    hb2m[off] = ab2m[r];
  }
}

// ---------------------------------------------------------------------------
// Pass 2: vertical 9-tap box, then b_cov / b2_cov epilogue.
// Out(16x16) = Band(16x24) x H(24x16); Band[i][t] = 1 iff i <= t <= i+8.
// ---------------------------------------------------------------------------
__global__ __launch_bounds__(256) void vbox_kernel(
    const float* __restrict__ hm, const float* __restrict__ hbm,
    const float* __restrict__ hb2m, const float* __restrict__ I,
    float* __restrict__ bcov, float* __restrict__ b2cov) {
  const int wave = threadIdx.x >> 5;
  const int lane = threadIdx.x & 31;
  const long tile = (long)blockIdx.x * 8 + wave;
  const int plane = (int)(tile >> 12);
  const int trem = (int)(tile & 4095);
  const int y0 = (trem >> 6) * 16;
  const int x0 = (trem & 63) * 16;
  const long pbase = (long)plane * NPLANE;

  const int arow = lane & 15;        // A: M index (output row i)
  const int khalf = (lane >> 4) * 2; // K base within chunk
  const int coln = lane & 15;        // B: N index

  v8f aKb = {}, aSb = {}, aSb2 = {};
#pragma unroll
  for (int c = 0; c < 6; ++c) {
    v2f aw, b_m, b_bm, b_b2;
#pragma unroll
    for (int s = 0; s < 2; ++s) {
      const int t = 4 * c + khalf + s;
      aw[s] = (t >= arow && t <= arow + 8) ? 1.f : 0.f;
      const int y = y0 - 4 + 4 * c + khalf + s;
      float m = 0.f, bm = 0.f, b2 = 0.f;
      if (y >= 0 && y < HH) {
        const long off = pbase + (long)y * WW + (x0 + coln);
        m = hm[off];
        bm = hbm[off];
        b2 = hb2m[off];
      }
      b_m[s] = m;
      b_bm[s] = bm;
      b_b2[s] = b2;
    }
    aKb = wmma4(aw, b_m, aKb);
    aSb = wmma4(aw, b_bm, aSb);
    aSb2 = wmma4(aw, b_b2, aSb2);
  }

  const int rofs = (lane >> 4) * 8;
#pragma unroll
  for (int r = 0; r < 8; ++r) {
    const long off = pbase + (long)(y0 + r + rofs) * WW + (x0 + coln);
    const float iv = I[off];
    const float m = iv > 0.f ? 1.f : 0.f;
    const float inv = 1.f / (aKb[r] + EPSF);
    float bc = aSb[r] * inv;
    bc = (m != 0.f) ? bc : 1.f; // b_cov*mask + (1-mask)
    bcov[off] = bc;
    b2cov[off] = aSb2[r] * inv;
  }
}

// ---------------------------------------------------------------------------
// Pass 3: per-chunk partial sums for class centers.
// grid = NB*CHUNKS blocks; block (b,g) covers PIX pixels, emits 12 floats:
// [num_c, den_c] for c = 0..5.
// ---------------------------------------------------------------------------
__global__ __launch_bounds__(256) void centers_kernel(
    const float* __restrict__ I, const float* __restrict__ U,
    const float* __restrict__ bcov, const float* __restrict__ b2cov,
    const int* __restrict__ p_ptr, float* __restrict__ part) {
  const int b_ = blockIdx.x >> 8;
  const int g = blockIdx.x & 255;
  const long base = (long)b_ * NPLANE + (long)g * PIX;
  const float p = (float)(*p_ptr);
  const bool p2 = (p == 2.f);

  float acc[12];
#pragma unroll
  for (int k = 0; k < 12; ++k) acc[k] = 0.f;

  for (int i = threadIdx.x; i < PIX; i += 256) {
    const long off = base + i;
    const float iv = I[off];
    if (iv > 0.f) {
      const float ib = iv * bcov[off];
      const float b2c = b2cov[off];
#pragma unroll
      for (int c = 0; c < NC; ++c) {
        const float uv = U[((long)(b_ * NC + c)) * NPLANE + (long)g * PIX + i];
        const float up = p2 ? uv * uv : powf(uv, p);
        acc[2 * c] += up * ib;
        acc[2 * c + 1] += up * b2c;
      }
    }
  }

  __shared__ float red[12 * 256];
#pragma unroll
  for (int k = 0; k < 12; ++k) red[k * 256 + threadIdx.x] = acc[k];
  __syncthreads();
  if (threadIdx.x < 12) {
    float s = 0.f;
    for (int t = 0; t < 256; ++t) s += red[threadIdx.x * 256 + t]; // fixed order
    part[(long)blockIdx.x * 12 + threadIdx.x] = s;
  }
}

// ---------------------------------------------------------------------------
// Pass 4: fold chunk partials into v[b][c] = num / (den + eps).
// ---------------------------------------------------------------------------
__global__ __launch_bounds__(64) void reduce_centers_kernel(
    const float* __restrict__ part, float* __restrict__ vv) {
  __shared__ float sums[NB * 12];
  const int j = threadIdx.x;
  if (j < NB * 12) {
    const int b_ = j / 12, slot = j % 12;
    float s = 0.f;
    for (int g = 0; g < CHUNKS; ++g)
      s += part[((long)(b_ * CHUNKS + g)) * 12 + slot];
    sums[j] = s;
  }
  __syncthreads();
  if (j < NB * NC) {
    const int b_ = j / NC, c = j % NC;
    vv[j] = sums[b_ * 12 + 2 * c] / (sums[b_ * 12 + 2 * c + 1] + EPSF);
  }
}

// ---------------------------------------------------------------------------
// Pass 5: membership update + squared-diff partial sums per chunk.
// ---------------------------------------------------------------------------
__global__ __launch_bounds__(256) void loss_kernel(
    const float* __restrict__ I, const float* __restrict__ U,
    const float* __restrict__ bcov, const float* __restrict__ vv,
    const int* __restrict__ p_ptr, float* __restrict__ lpart) {
  const int b_ = blockIdx.x >> 8;
  const int g = blockIdx.x & 255;
  const long base = (long)b_ * NPLANE + (long)g * PIX;
  const float p = (float)(*p_ptr);
  const bool p2 = (p == 2.f);
  const float q = 1.f / (p - 1.f);

  float vc[NC];
#pragma unroll
  for (int c = 0; c < NC; ++c) vc[c] = vv[b_ * NC + c];

  float acc = 0.f;
  for (int i = threadIdx.x; i < PIX; i += 256) {
    const long off = base + i;
    const float iv = I[off];
    const float bc = bcov[off];
    float D[NC], fs = 0.f;
#pragma unroll
    for (int c = 0; c < NC; ++c) {
      const float r = iv - vc[c] * bc;
      const float rr = r * r;
      const float Dc = (p2 ? rr : powf(rr, q)) + EPSF;
      D[c] = Dc;
      fs += 1.f / Dc;
    }
#pragma unroll
    for (int c = 0; c < NC; ++c) {
      const float nu = 1.f / (D[c] * fs + EPSF);
      const float uv = U[((long)(b_ * NC + c)) * NPLANE + (long)g * PIX + i];
      const float d = uv - nu;
      acc += d * d;
    }
  }

  __shared__ float red[256];
  red[threadIdx.x] = acc;
  __syncthreads();
  for (int s = 128; s > 0; s >>= 1) {
    if (threadIdx.x < s) red[threadIdx.x] += red[threadIdx.x + s];
    __syncthreads();
  }
  if (threadIdx.x == 0) lpart[blockIdx.x] = red[0];
}

// ---------------------------------------------------------------------------
// Pass 6: final scalar mean.
// ---------------------------------------------------------------------------
__global__ __launch_bounds__(256) void final_kernel(
    const float* __restrict__ lpart, float* __restrict__ out) {
  __shared__ float red[256];
  float a = 0.f;
#pragma unroll
  for (int k = 0; k < 4; ++k) a += lpart[threadIdx.x + 256 * k];
  red[threadIdx.x] = a;
  __syncthreads();
  for (int s = 128; s > 0; s >>= 1) {
    if (threadIdx.x < s) red[threadIdx.x] += red[threadIdx.x + s];
    __syncthreads();
  }
  if (threadIdx.x == 0)
    out[0] = red[0] * (1.f / ((float)NB * NC * NPLANE));
}

extern "C" void kernel_launch(void* const* d_in, const int* in_sizes, int n_in,
                              void* d_out, int out_size, void* d_ws,
                              size_t ws_size, hipStream_t stream) {
  (void)in_sizes; (void)n_in; (void)out_size; (void)ws_size;
  const float* I = (const float*)d_in[0]; // [4,1,1024,1024]
  const float* U = (const float*)d_in[1]; // [4,6,1024,1024]
  const float* B = (const float*)d_in[2]; // [4,1,1024,1024]
  const int* p = (const int*)d_in[3];     // scalar p (sigma=2 baked into K=9)

  float* ws = (float*)d_ws;
  const size_t P = (size_t)NB * NPLANE; // 4M floats per field
  float* hm = ws;
  float* hbm = ws + P;
  float* hb2m = ws + 2 * P;
  float* bcov = ws + 3 * P;
  float* b2cov = ws + 4 * P;
  float* cpart = ws + 5 * P;                 // NB*CHUNKS*12
  float* vv = cpart + (size_t)NB * CHUNKS * 12; // NB*NC
  float* lpart = vv + NB * NC;               // NB*CHUNKS

  hbox_kernel<<<2048, 256, 0, stream>>>(I, B, hm, hbm, hb2m);
  vbox_kernel<<<2048, 256, 0, stream>>>(hm, hbm, hb2m, I, bcov, b2cov);
  centers_kernel<<<NB * CHUNKS, 256, 0, stream>>>(I, U, bcov, b2cov, p, cpart);
  reduce_centers_kernel<<<1, 64, 0, stream>>>(cpart, vv);
  loss_kernel<<<NB * CHUNKS, 256, 0, stream>>>(I, U, bcov, vv, p, lpart);
  final_kernel<<<1, 256, 0, stream>>>(lpart, (float*)d_out);
}